// SageConv_22170621182315
// MI455X (gfx1250) — compile-verified
//
#include <hip/hip_runtime.h>

// Problem constants (match reference setup_inputs)
#define N_SRC   131072
#define N_DST   65536
#define N_EDGES 524288
#define D       128     // D_IN == D_OUT == 128

typedef __attribute__((ext_vector_type(2))) float v2f;
typedef __attribute__((ext_vector_type(4))) float v4f;
typedef __attribute__((ext_vector_type(8))) float v8f;

#define WMMA_F32(a, b, c) \
    __builtin_amdgcn_wmma_f32_16x16x4_f32(false, (a), false, (b), (short)0, (c), false, false)

// ---------------------------------------------------------------------------
// Kernel 1: segment-mean aggregation (the HBM-bound phase).
// One wave32 per destination row. dst_idx is sorted -> binary search gives
// the contiguous edge range [start,end). Each lane owns 4 columns (float4),
// so a gathered feat row is one fully coalesced 512B wave request.
// Unrolled x2 with dual accumulators for memory-level parallelism.
// ---------------------------------------------------------------------------
__global__ __launch_bounds__(256) void sage_neigh_mean_kernel(
    const float* __restrict__ feat,
    const int*   __restrict__ src_idx,
    const int*   __restrict__ dst_idx,
    float*       __restrict__ h_neigh)
{
    const int wave = threadIdx.x >> 5;
    const int lane = threadIdx.x & 31;
    const int row  = blockIdx.x * 8 + wave;
    if (row >= N_DST) return;

    // lower_bound(dst_idx, row)
    int lo = 0, hi = N_EDGES;
    while (lo < hi) {
        int mid = (lo + hi) >> 1;
        if (dst_idx[mid] < row) lo = mid + 1; else hi = mid;
    }
    const int start = lo;
    // lower_bound(dst_idx, row + 1)
    hi = N_EDGES;
    while (lo < hi) {
        int mid = (lo + hi) >> 1;
        if (dst_idx[mid] < row + 1) lo = mid + 1; else hi = mid;
    }
    const int end = lo;

    const int col4 = lane * 4;
    v4f acc0 = {0.0f, 0.0f, 0.0f, 0.0f};
    v4f acc1 = {0.0f, 0.0f, 0.0f, 0.0f};

    int e = start;
    for (; e + 1 < end; e += 2) {
        const int s0 = src_idx[e];
        const int s1 = src_idx[e + 1];
        acc0 += *(const v4f*)(feat + (size_t)s0 * D + col4);
        acc1 += *(const v4f*)(feat + (size_t)s1 * D + col4);
    }
    if (e < end) {
        const int s0 = src_idx[e];
        acc0 += *(const v4f*)(feat + (size_t)s0 * D + col4);
    }
    acc0 += acc1;

    const int deg = end - start;
    const float inv = 1.0f / (float)(deg > 1 ? deg : 1);
    acc0 *= inv;

    *(v4f*)(h_neigh + (size_t)row * D + col4) = acc0;
}

// ---------------------------------------------------------------------------
// Kernel 2: fused dual GEMM + bias with V_WMMA_F32_16X16X4_F32 (fp32, exact
// numerics vs reference):
//   out = h_self @ W_self + h_neigh @ W_neigh + (b_self + b_neigh)
//
// Tiling: block = 256 threads (8 waves). Block tile = 64 rows x 128 cols.
// Each wave owns a 64-row x 16-col strip = four 16x16 accumulators that all
// share one B-fragment per K-step -> W is read from L2 only once per block
// (1024 blocks x 128KB = 128MB L2 traffic, 4x less than a 16-row tile).
// The four accumulators are interleaved (c0,c1,c2,c3) so dependent WMMAs are
// 3 instructions apart, covering the f32-WMMA D->C RAW latency.
//
// Fragment layouts (ISA 7.12.2, fp32 16x16x4):
//   A 16x4: lane<16 -> K=k,k+1 of row M=lane; lane>=16 -> K=k+2,k+3
//           (row-major activation => contiguous float2 load)
//   B 4x16: lane<16 -> B[k][n],B[k+1][n]; lane>=16 -> B[k+2][n],B[k+3][n]
//   C 16x16: VGPR r -> M=r (lanes 0-15) / M=r+8 (lanes 16-31), N=lane&15
// ---------------------------------------------------------------------------
__global__ __launch_bounds__(256) void sage_gemm_kernel(
    const float* __restrict__ h_self,    // feat[:N_DST]
    const float* __restrict__ h_neigh,
    const float* __restrict__ W_self,    // [D][D] row-major
    const float* __restrict__ W_neigh,   // [D][D] row-major
    const float* __restrict__ b_self,
    const float* __restrict__ b_neigh,
    float*       __restrict__ out)
{
    const int wave  = threadIdx.x >> 5;
    const int lane  = threadIdx.x & 31;
    const int m0    = blockIdx.x * 64;       // 64-row block tile
    const int n0    = wave * 16;             // 16-col wave strip
    const int l15   = lane & 15;
    const int hi16  = lane >> 4;             // 0 or 1

    const int n     = n0 + l15;              // B-frag / output column
    const int khalf = hi16 * 2;              // K offset within the 4-step

    const float* aS = h_self  + (size_t)(m0 + l15) * D;   // A rows, tile 0
    const float* aN = h_neigh + (size_t)(m0 + l15) * D;

    v8f c0 = {}, c1 = {}, c2 = {}, c3 = {};

    #pragma unroll 2
    for (int k = 0; k < D; k += 4) {
        const int kk = k + khalf;

        // One B-fragment per W, shared by all four row tiles.
        v2f bS, bN;
        bS.x = W_self [(size_t)kk * D + n];
        bS.y = W_self [(size_t)(kk + 1) * D + n];
        bN.x = W_neigh[(size_t)kk * D + n];
        bN.y = W_neigh[(size_t)(kk + 1) * D + n];

        v2f a0 = *(const v2f*)(aS + kk);
        v2f a1 = *(const v2f*)(aS + 16 * D + kk);
        v2f a2 = *(const v2f*)(aS + 32 * D + kk);
        v2f a3 = *(const v2f*)(aS + 48 * D + kk);
        c0 = WMMA_F32(a0, bS, c0);
        c1 = WMMA_F32(a1, bS, c1);
        c2 = WMMA_F32(a2, bS, c2);
        c3 = WMMA_F32(a3, bS, c3);

        a0 = *(const v2f*)(aN + kk);
        a1 = *(const v2f*)(aN + 16 * D + kk);
        a2 = *(const v2f*)(aN + 32 * D + kk);
        a3 = *(const v2f*)(aN + 48 * D + kk);
        c0 = WMMA_F32(a0, bN, c0);
        c1 = WMMA_F32(a1, bN, c1);
        c2 = WMMA_F32(a2, bN, c2);
        c3 = WMMA_F32(a3, bN, c3);
    }

    const float bias = b_self[n] + b_neigh[n];
    const int mbase = m0 + hi16 * 8;         // C-layout: lanes>=16 hold M=r+8
    #pragma unroll
    for (int r = 0; r < 8; ++r) {
        out[(size_t)(mbase +      r) * D + n] = c0[r] + bias;
        out[(size_t)(mbase + 16 + r) * D + n] = c1[r] + bias;
        out[(size_t)(mbase + 32 + r) * D + n] = c2[r] + bias;
        out[(size_t)(mbase + 48 + r) * D + n] = c3[r] + bias;
    }
}

// ---------------------------------------------------------------------------
// Host launch. Input order (setup_inputs dict):
//   0: feat [N_SRC*D] f32      1: W_self [D*D] f32   2: b_self [D] f32
//   3: W_neigh [D*D] f32       4: b_neigh [D] f32
//   5: src_idx [N_EDGES] i32   6: dst_idx [N_EDGES] i32 (sorted)
//   7: num_row (scalar i32, == N_DST)
// Output: [N_DST*D] f32. Workspace: h_neigh (N_DST*D f32 = 33.5 MB).
// ---------------------------------------------------------------------------
extern "C" void kernel_launch(void* const* d_in, const int* in_sizes, int n_in,
                              void* d_out, int out_size, void* d_ws, size_t ws_size,
                              hipStream_t stream) {
    const float* feat    = (const float*)d_in[0];
    const float* W_self  = (const float*)d_in[1];
    const float* b_self  = (const float*)d_in[2];
    const float* W_neigh = (const float*)d_in[3];
    const float* b_neigh = (const float*)d_in[4];
    const int*   src_idx = (const int*)d_in[5];
    const int*   dst_idx = (const int*)d_in[6];

    float* out     = (float*)d_out;
    float* h_neigh = (float*)d_ws;   // N_DST * D floats

    // Phase 1: segment-mean -> h_neigh (8 rows per 256-thread block)
    sage_neigh_mean_kernel<<<N_DST / 8, 256, 0, stream>>>(
        feat, src_idx, dst_idx, h_neigh);

    // Phase 2: fused dual fp32-WMMA GEMM + bias (64 rows per block)
    sage_gemm_kernel<<<N_DST / 64, 256, 0, stream>>>(
        feat, h_neigh, W_self, W_neigh, b_self, b_neigh, out);
}